// RelGraphConvLayer_67001489817705
// MI455X (gfx1250) — compile-verified
//
#include <hip/hip_runtime.h>
#include <hip/hip_bf16.h>

// ---------------------------------------------------------------------------
// RelGraphConv forward for MI455X (gfx1250, wave32).
//   Phase 0: transpose all relation weights into d_ws (Wt[r][n][k]) so the
//            WMMA B-fragment becomes one global_load_b64 per lane.
//   Phase 1 (per relation): zero accumulator+degree, scatter-add feat rows
//                           over edges with f32 global atomics (L2-resident).
//   Phase 2 (per relation): normalized-A @ W_r via V_WMMA_F32_16X16X4_F32,
//                           accumulated into d_out (bias folded in on r==0).
// ---------------------------------------------------------------------------

typedef __attribute__((ext_vector_type(2))) float v2f;
typedef __attribute__((ext_vector_type(8))) float v8f;

#define DIM 256            // IN_DIM == OUT_DIM == 256
#define LDS_STRIDE 260     // padded row stride: (260 % 64) = 4 -> conflict-free

// ---- zero fill -------------------------------------------------------------
__global__ __launch_bounds__(256) void rgc_zero_f32(float* __restrict__ p, int n) {
    int i = blockIdx.x * blockDim.x + threadIdx.x;
    if (i < n) p[i] = 0.0f;
}

// ---- weight transpose: Wt[r][n][k] = W[r][k][n] ----------------------------
// LDS-tiled 32x32 transpose, coalesced on both sides. Grid: (DIM/32, DIM/32, R)
__global__ __launch_bounds__(256) void rgc_transpose_w(const float* __restrict__ W,
                                                       float* __restrict__ Wt) {
    __shared__ float tile[32][33];   // +1 pad: conflict-free
    const int tx = threadIdx.x & 31;
    const int ty = threadIdx.x >> 5; // 0..7
    const int k0 = blockIdx.x * 32;
    const int n0 = blockIdx.y * 32;
    const float* Wr  = W  + (size_t)blockIdx.z * DIM * DIM;
    float*       Wtr = Wt + (size_t)blockIdx.z * DIM * DIM;
    #pragma unroll
    for (int j = 0; j < 32; j += 8)
        tile[ty + j][tx] = Wr[(size_t)(k0 + ty + j) * DIM + n0 + tx];
    __syncthreads();
    #pragma unroll
    for (int j = 0; j < 32; j += 8)
        Wtr[(size_t)(n0 + ty + j) * DIM + k0 + tx] = tile[tx][ty + j];
}

// ---- edge scatter: one wave32 per edge ------------------------------------
// lane handles float4 chunks [lane] and [lane+32] of the 256-float row.
__global__ __launch_bounds__(256) void rgc_scatter(const float* __restrict__ feat,
                                                   const int* __restrict__ src,
                                                   const int* __restrict__ dst,
                                                   float* __restrict__ agg,
                                                   float* __restrict__ deg,
                                                   int n_edges) {
    int gtid = blockIdx.x * blockDim.x + threadIdx.x;
    int edge = gtid >> 5;
    int lane = gtid & 31;
    if (edge >= n_edges) return;

    int u = src[edge];
    int v = dst[edge];

    const float4* fr = (const float4*)(feat + (size_t)u * DIM);
    float4 x0 = fr[lane];
    float4 x1 = fr[lane + 32];

    float* a = agg + (size_t)v * DIM;
    int o0 = lane * 4;
    int o1 = (lane + 32) * 4;
    atomicAdd(&a[o0 + 0], x0.x);
    atomicAdd(&a[o0 + 1], x0.y);
    atomicAdd(&a[o0 + 2], x0.z);
    atomicAdd(&a[o0 + 3], x0.w);
    atomicAdd(&a[o1 + 0], x1.x);
    atomicAdd(&a[o1 + 1], x1.y);
    atomicAdd(&a[o1 + 2], x1.z);
    atomicAdd(&a[o1 + 3], x1.w);
    if (lane == 0) atomicAdd(&deg[v], 1.0f);
}

// ---- fused normalize + GEMM tile kernel -----------------------------------
// Block = 256 threads = 8 waves. blockIdx.x -> 16-row tile, (blockIdx.y*8+wave)
// -> 16-col tile. A-tile (normalized) staged in LDS, shared by all 8 waves.
// B read from transposed weights: lane loads Wt[col][ka..ka+1] as one b64.
template <bool FIRST>
__global__ __launch_bounds__(256) void rgc_gemm_wmma(const float* __restrict__ agg,
                                                     const float* __restrict__ deg,
                                                     const float* __restrict__ Wt,
                                                     const float* __restrict__ bias,
                                                     float* __restrict__ out) {
    __shared__ float As[16 * LDS_STRIDE];
    __shared__ float rdS[16];

    const int tid  = threadIdx.x;
    const int wave = tid >> 5;
    const int lane = tid & 31;
    const int half = lane >> 4;   // 0: lanes 0-15, 1: lanes 16-31
    const int l16  = lane & 15;

    const int row0 = blockIdx.x * 16;
    const int col0 = (blockIdx.y * 8 + wave) * 16;

    // reciprocal of clamped in-degree (norm='right')
    if (tid < 16) {
        float d = deg[row0 + tid];
        rdS[tid] = 1.0f / fmaxf(d, 1.0f);
    }
    __syncthreads();

    // cooperative coalesced fill of normalized A tile: thread tid -> column tid
    #pragma unroll
    for (int j = 0; j < 16; ++j) {
        As[j * LDS_STRIDE + tid] = agg[(size_t)(row0 + j) * DIM + tid] * rdS[j];
    }
    __syncthreads();

    // lane's B column, contiguous in K via transposed weights
    const int col = col0 + l16;
    const float* Bcol = Wt + (size_t)col * DIM;

    // K-loop: 64 x V_WMMA_F32_16X16X4_F32 (D = A x B + C, f32 throughout)
    v8f c = {};
    #pragma unroll 4
    for (int k = 0; k < DIM; k += 4) {
        const int ka = k + 2 * half;  // lanes 0-15: K=k,k+1; 16-31: K=k+2,k+3
        v2f a, b;
        a.x = As[l16 * LDS_STRIDE + ka];
        a.y = As[l16 * LDS_STRIDE + ka + 1];
        b = *(const v2f*)(Bcol + ka);            // global_load_b64
        c = __builtin_amdgcn_wmma_f32_16x16x4_f32(
                /*neg_a=*/false, a, /*neg_b=*/false, b,
                /*c_mod=*/(short)0, c, /*reuse_a=*/false, /*reuse_b=*/false);
    }

    // C/D layout: VGPR v -> row (row0 + v + 8*half), col (col0 + l16)
    float* O = out + (size_t)(row0 + 8 * half) * DIM + col;
    if (FIRST) {
        const float bv = bias[col];
        #pragma unroll
        for (int v = 0; v < 8; ++v)
            O[(size_t)v * DIM] = c[v] + bv;
    } else {
        float prev[8];
        #pragma unroll
        for (int v = 0; v < 8; ++v)
            prev[v] = O[(size_t)v * DIM];
        #pragma unroll
        for (int v = 0; v < 8; ++v)
            O[(size_t)v * DIM] = prev[v] + c[v];
    }
}

// ---------------------------------------------------------------------------
extern "C" void kernel_launch(void* const* d_in, const int* in_sizes, int n_in,
                              void* d_out, int out_size, void* d_ws, size_t ws_size,
                              hipStream_t stream) {
    const float* feat   = (const float*)d_in[0];
    const float* weight = (const float*)d_in[1];
    const float* bias   = (const float*)d_in[2];
    const int*   e_src  = (const int*)d_in[3];
    const int*   e_dst  = (const int*)d_in[4];
    float*       out    = (float*)d_out;

    const int N = in_sizes[0] / DIM;           // 50000 nodes
    const int R = in_sizes[1] / (DIM * DIM);   // 4 relations
    const int E = in_sizes[3] / R;             // 1,000,000 edges per relation

    float* agg = (float*)d_ws;                    // [N, 256] f32 accumulator
    float* deg = agg + (size_t)N * DIM;           // [N] f32 degree
    float* Wt  = deg + N;                         // [R, 256, 256] transposed W

    const int aggN = N * DIM;
    const dim3 zeroAGrid((aggN + 255) / 256);
    const dim3 zeroDGrid((N + 255) / 256);
    const dim3 scatGrid((E * 32 + 255) / 256);
    const dim3 gemmGrid(N / 16, (DIM / 16) / 8);   // 3125 x 2 tile blocks
    const dim3 trGrid(DIM / 32, DIM / 32, R);

    rgc_transpose_w<<<trGrid, 256, 0, stream>>>(weight, Wt);

    for (int r = 0; r < R; ++r) {
        rgc_zero_f32<<<zeroAGrid, 256, 0, stream>>>(agg, aggN);
        rgc_zero_f32<<<zeroDGrid, 256, 0, stream>>>(deg, N);
        rgc_scatter<<<scatGrid, 256, 0, stream>>>(feat,
                                                  e_src + (size_t)r * E,
                                                  e_dst + (size_t)r * E,
                                                  agg, deg, E);
        const float* Wtr = Wt + (size_t)r * DIM * DIM;
        if (r == 0)
            rgc_gemm_wmma<true><<<gemmGrid, 256, 0, stream>>>(agg, deg, Wtr, bias, out);
        else
            rgc_gemm_wmma<false><<<gemmGrid, 256, 0, stream>>>(agg, deg, Wtr, bias, out);
    }
}